// YoloLoss_68006512164891
// MI455X (gfx1250) — compile-verified
//
#include <hip/hip_runtime.h>
#include <math.h>

// ---------------------------------------------------------------------------
// YOLO loss, MI455X (gfx1250, wave32).
// Bandwidth-bound streaming reduction: ~177 MB read / launch -> ~7.6 us at
// 23.3 TB/s. One cell per lane, float2/float4 vector loads, wave32 shuffle
// reduction, per-block V_WMMA_F32_16X16X4_F32 (ones-A column-sum trick) for
// the cross-wave partial reduction, single f32 atomic per block.
// ---------------------------------------------------------------------------

typedef __attribute__((ext_vector_type(2))) float v2f;
typedef __attribute__((ext_vector_type(8))) float v8f;

constexpr int   kS   = 14;
constexpr int   kB   = 2;
constexpr int   kNC  = 20;
constexpr int   kCH  = kB * 5 + kNC;   // 30 channels per cell
constexpr float kInvS = 1.0f / 14.0f;

__device__ __forceinline__ float clamp01(float x) {
    return fminf(fmaxf(x, 0.0f), 1.0f);
}

__global__ void yolo_zero_out(float* out) {
    if (threadIdx.x == 0) out[0] = 0.0f;
}

__global__ __launch_bounds__(256) void yolo_loss_kernel(
    const float* __restrict__ pred,     // [cells, 30]
    const float* __restrict__ tbox,     // [cells, 4]
    const float* __restrict__ tcls,     // [cells, 20]
    const int*   __restrict__ objmap,   // [cells]
    float*       __restrict__ out,      // [1]
    int cells, float inv_n)
{
    __shared__ float wsum[8];

    float acc = 0.0f;
    const int tid    = blockIdx.x * blockDim.x + threadIdx.x;
    const int stride = gridDim.x * blockDim.x;

    for (int cell = tid; cell < cells; cell += stride) {
        // gfx1250 prefetch path (global_prefetch_b8) for the next stride.
        const int nxt = cell + stride;
        if (nxt < cells) {
            __builtin_prefetch(pred + (size_t)nxt * kCH, 0, 1);
            __builtin_prefetch(tcls + (size_t)nxt * kNC, 0, 1);
        }

        // ---- loads (cell stride 120 B -> 8B aligned; use b64 loads) ----
        float p[kCH];
        const float2* pp = reinterpret_cast<const float2*>(pred + (size_t)cell * kCH);
        #pragma unroll
        for (int i = 0; i < kCH / 2; ++i) {
            float2 v = pp[i];
            p[2 * i] = v.x; p[2 * i + 1] = v.y;
        }
        const float4 tb = reinterpret_cast<const float4*>(tbox)[cell];
        float tc[kNC];
        const float4* tcp = reinterpret_cast<const float4*>(tcls + (size_t)cell * kNC);
        #pragma unroll
        for (int i = 0; i < kNC / 4; ++i) {
            float4 v = tcp[i];
            tc[4 * i] = v.x; tc[4 * i + 1] = v.y; tc[4 * i + 2] = v.z; tc[4 * i + 3] = v.w;
        }
        const float obj = objmap[cell] ? 1.0f : 0.0f;

        // ---- class BCE (note: 1-pc is exact in f32 for pc in [0.5,1],
        //      so logf(1-pc) == log1pf(-pc) to log accuracy) ----
        float bce = 0.0f;
        #pragma unroll
        for (int j = 0; j < kNC; ++j) {
            float pc = fminf(fmaxf(p[10 + j], 1e-7f), 1.0f - 1e-7f);
            bce -= tc[j] * __logf(pc) + (1.0f - tc[j]) * __logf(1.0f - pc);
        }

        // ---- no-object confidence loss ----
        const float c0 = p[4] - 0.05f;
        const float c1 = p[9] - 0.05f;
        const float noobj = (1.0f - obj) * (c0 * c0 + c1 * c1);

        // ---- target box -> xyxy ----
        const float tx = tb.x * kInvS, ty = tb.y * kInvS;
        const float tw = fmaxf(tb.z, 1e-10f), th = fmaxf(tb.w, 1e-10f);
        const float tx1 = clamp01(tx - 0.5f * tw), ty1 = clamp01(ty - 0.5f * th);
        const float tx2 = clamp01(tx + 0.5f * tw), ty2 = clamp01(ty + 0.5f * th);
        const float t_area = (tx2 - tx1) * (ty2 - ty1);

        // ---- per-box IoU ----
        float ious[2];
        #pragma unroll
        for (int b = 0; b < 2; ++b) {
            const float bx = p[b * 5 + 0] * kInvS, by = p[b * 5 + 1] * kInvS;
            const float bw = fmaxf(p[b * 5 + 2], 1e-10f), bh = fmaxf(p[b * 5 + 3], 1e-10f);
            const float x1 = clamp01(bx - 0.5f * bw), y1 = clamp01(by - 0.5f * bh);
            const float x2 = clamp01(bx + 0.5f * bw), y2 = clamp01(by + 0.5f * bh);
            const float lx = fmaxf(x1, tx1), ly = fmaxf(y1, ty1);
            const float rx = fminf(x2, tx2), ry = fminf(y2, ty2);
            const float iw = fmaxf(rx - lx, 0.0f), ih = fmaxf(ry - ly, 0.0f);
            const float inter = iw * ih;
            const float pa = (x2 - x1) * (y2 - y1);
            ious[b] = inter / (pa + t_area - inter + 1e-10f);
        }
        const int   bi       = (ious[1] > ious[0]) ? 1 : 0;   // argmax: first on tie
        const float best_iou = fminf(fmaxf(ious[0], ious[1]) + 0.5f, 0.95f);

        const float bb0 = p[bi * 5 + 0], bb1 = p[bi * 5 + 1];
        const float bb2 = p[bi * 5 + 2], bb3 = p[bi * 5 + 3];
        const float bb4 = p[bi * 5 + 4];

        // ---- regression + containment ----
        const float pw  = sqrtf(fabsf(bb2) + 1e-6f);
        const float ph  = sqrtf(fabsf(bb3) + 1e-6f);
        const float tws = sqrtf(fabsf(tb.z) + 1e-6f);
        const float ths = sqrtf(fabsf(tb.w) + 1e-6f);
        const float dx = bb0 - tb.x, dy = bb1 - tb.y;
        const float dw = pw - tws,   dh = ph - ths;
        const float reg     = dx * dx + dy * dy + dw * dw + dh * dh;
        const float contain = (bb4 - best_iou) * (bb4 - best_iou) * 10.0f;

        // CLASS_W=0.5, L_NOOBJ=0.5, COORD_SCALE*L_COORD=10, BOOST=10
        acc += 0.5f * bce * obj + 0.5f * noobj + 10.0f * reg * obj + contain * obj;
    }

    // ---- intra-wave reduction (wave32) ----
    #pragma unroll
    for (int off = 16; off > 0; off >>= 1)
        acc += __shfl_down(acc, off, 32);

    const int lane = threadIdx.x & 31;
    const int wave = threadIdx.x >> 5;
    if (lane == 0) wsum[wave] = acc;
    __syncthreads();

    // ---- cross-wave reduction on wave 0 via V_WMMA_F32_16X16X4_F32 ----
    // A = ones(16x4)  =>  D[m][n] = sum_k B[k][n]  (column sums).
    // Sum over row 0 of D == grand sum of all 64 B elements, independent of
    // the exact B lane layout. B carries the 8 wave partials + zeros.
    // Whole wave takes the branch -> EXEC all 1s as WMMA requires.
    if (threadIdx.x < 32) {
        const float part = (threadIdx.x < 8) ? wsum[threadIdx.x] : 0.0f;
        v2f a; a[0] = 1.0f; a[1] = 1.0f;          // all-ones A
        v2f b; b[0] = part; b[1] = 0.0f;          // partials somewhere in B
        v8f c = {};
        c = __builtin_amdgcn_wmma_f32_16x16x4_f32(
                false, a, false, b, (short)0, c, false, false);
        // Row 0 of D lives in c[0] across lanes 0..15 (lanes 16..31 hold row 8
        // — also column sums; mask them out and reduce the 16 row-0 entries).
        float r = (threadIdx.x < 16) ? c[0] : 0.0f;
        #pragma unroll
        for (int off = 8; off > 0; off >>= 1)
            r += __shfl_down(r, off, 32);
        if (threadIdx.x == 0)
            atomicAdd(out, r * inv_n);
    }
}

extern "C" void kernel_launch(void* const* d_in, const int* in_sizes, int n_in,
                              void* d_out, int out_size, void* d_ws, size_t ws_size,
                              hipStream_t stream) {
    const float* pred   = (const float*)d_in[0];   // [N,S,S,30]
    const float* tbox   = (const float*)d_in[1];   // [N,S,S,4]
    const float* tcls   = (const float*)d_in[2];   // [N,S,S,20]
    const int*   objmap = (const int*)d_in[3];     // [N,S,S] (bool -> int per harness)
    float*       out    = (float*)d_out;

    const int cells = in_sizes[3];                 // N*S*S
    const int n     = cells / (kS * kS);
    const float inv_n = 1.0f / (float)n;

    yolo_zero_out<<<1, 32, 0, stream>>>(out);

    const int threads = 256;
    const int blocks  = (cells + threads - 1) / threads;   // 3136 for N=4096
    yolo_loss_kernel<<<blocks, threads, 0, stream>>>(
        pred, tbox, tcls, objmap, out, cells, inv_n);
}